// DirectVolumeRenderer_29403346108458
// MI455X (gfx1250) — compile-verified
//
#include <hip/hip_runtime.h>
#include <math.h>

typedef __attribute__((ext_vector_type(2))) float v2f;
typedef __attribute__((ext_vector_type(8))) float v8f;
// 4-byte-aligned float pair for x-adjacent corner loads (global_load_b64)
typedef float v2f_u __attribute__((ext_vector_type(2), aligned(4)));

#define NPIX  16384   // 128 x 128 rays
#define NPTS  256
#define SPL   8       // samples per lane (256 / 32 lanes)

// fx,fy,fz are voxel-space coords (align_corners grid). Zero padding outside.
// Samples both volumes at once with 8 paired (b64) gathers instead of 16.
__device__ __forceinline__ void tri_sample(const float* __restrict__ img,
                                           const float* __restrict__ opa,
                                           float fx, float fy, float fz,
                                           float& feat, float& dens)
{
    feat = 0.f; dens = 0.f;
    // completely outside -> every corner invalid -> contribution 0
    if (fx < -1.f || fx >= 128.f || fy < -1.f || fy >= 128.f ||
        fz < -1.f || fz >= 128.f)
        return;

    float xf = floorf(fx), yf = floorf(fy), zf = floorf(fz);
    int x0 = (int)xf, y0 = (int)yf, z0 = (int)zf;       // in [-1,127]
    float wx1 = fx - xf, wy1 = fy - yf, wz1 = fz - zf;
    float wx0 = 1.f - wx1, wy0 = 1.f - wy1, wz0 = 1.f - wz1;

    // ---- x axis: one contiguous pair load at xb, weights remapped to slots.
    // interior: (wx0, wx1); x0==-1 -> slot0 is x=0=x0+1: (wx1, 0);
    // x0==127 -> slot1 is x=127=x0: (0, wx0). Masks absorbed.
    const int xb = min(max(x0, 0), 126);
    const float wA = (x0 == xb) ? wx0 : ((x0 + 1 == xb) ? wx1 : 0.f);
    const float wB = (x0 == xb) ? wx1 : ((x0 == xb + 1) ? wx0 : 0.f);

    // ---- y/z axes: fold validity masks into axis weights, clamp indices
    wy0 *= (y0 >= 0)   ? 1.f : 0.f;
    wy1 *= (y0 <= 126) ? 1.f : 0.f;
    wz0 *= (z0 >= 0)   ? 1.f : 0.f;
    wz1 *= (z0 <= 126) ? 1.f : 0.f;
    const int yc0 = max(y0, 0), yc1 = min(y0 + 1, 127);
    const int zc0 = max(z0, 0), zc1 = min(z0 + 1, 127);

    const int zb0 = zc0 << 14, zb1 = zc1 << 14;   // z * 128 * 128
    const int yb0 = yc0 << 7,  yb1 = yc1 << 7;    // y * 128

    const int b00 = zb0 + yb0 + xb, b01 = zb0 + yb1 + xb;
    const int b10 = zb1 + yb0 + xb, b11 = zb1 + yb1 + xb;

    // 8 paired gathers (4 per volume), issued back-to-back
    const v2f_u f00 = *(const v2f_u*)(img + b00);
    const v2f_u f01 = *(const v2f_u*)(img + b01);
    const v2f_u f10 = *(const v2f_u*)(img + b10);
    const v2f_u f11 = *(const v2f_u*)(img + b11);
    const v2f_u d00 = *(const v2f_u*)(opa + b00);
    const v2f_u d01 = *(const v2f_u*)(opa + b01);
    const v2f_u d10 = *(const v2f_u*)(opa + b10);
    const v2f_u d11 = *(const v2f_u*)(opa + b11);

    const float w00 = wz0 * wy0, w01 = wz0 * wy1;
    const float w10 = wz1 * wy0, w11 = wz1 * wy1;

    // bilinear combine per x-slot, then blend the two x-slots
    const float fA = w00 * f00.x + w01 * f01.x + w10 * f10.x + w11 * f11.x;
    const float fB = w00 * f00.y + w01 * f01.y + w10 * f10.y + w11 * f11.y;
    const float dA = w00 * d00.x + w01 * d01.x + w10 * d10.x + w11 * d11.x;
    const float dB = w00 * d00.y + w01 * d01.y + w10 * d10.y + w11 * d11.y;

    feat = wA * fA + wB * fB;
    dens = wA * dA + wB * dB;
}

// One wave32 per ray. Lane l handles samples [8l, 8l+8); transmittance is
// combined across lanes with a log-step shuffle scan (segmented cumprod).
__global__ void __launch_bounds__(256)
dvr_render_kernel(const float* __restrict__ img, const float* __restrict__ opa,
                  const float* __restrict__ R, const float* __restrict__ T,
                  float* __restrict__ gray)
{
    const int lane = threadIdx.x & 31;
    const int ray  = blockIdx.x * (blockDim.x >> 5) + (threadIdx.x >> 5);
    // output layout is [W,H] (after the (0,3,2,1) permute): ray = w*128 + h
    const int w = ray >> 7;
    const int h = ray & 127;

    const float inv_focal = 1.0f / 1.7320508f;
    const float gx = 1.f - (2.f / 127.f) * (float)w;   // xs[w]
    const float gy = 1.f - (2.f / 127.f) * (float)h;   // ys[h]
    const float dcx = gx * inv_focal, dcy = gy * inv_focal, dcz = 1.f;

    const float r00 = R[0], r01 = R[1], r02 = R[2];
    const float r10 = R[3], r11 = R[4], r12 = R[5];
    const float r20 = R[6], r21 = R[7], r22 = R[8];
    const float t0 = T[0], t1 = T[1], t2 = T[2];

    // origin_j = -sum_i R[j,i] * T_i ; dir_j = sum_i R[j,i] * dc_i
    const float ox = -(r00 * t0 + r01 * t1 + r02 * t2);
    const float oy = -(r10 * t0 + r11 * t1 + r12 * t2);
    const float oz = -(r20 * t0 + r21 * t1 + r22 * t2);
    const float dwx = r00 * dcx + r01 * dcy + r02 * dcz;
    const float dwy = r10 * dcx + r11 * dcy + r12 * dcz;
    const float dwz = r20 * dcx + r21 * dcy + r22 * dcz;

    // voxel coord = ((p/half_extent)+1)*63.5 ;  half_extent = (3/128)*127/2
    const float K = 63.5f / 1.48828125f;
    const float Ax = ox * K + 63.5f, Bx = dwx * K;
    const float Ay = oy * K + 63.5f, By = dwy * K;
    const float Az = oz * K + 63.5f, Bz = dwz * K;

    float Tl = 1.f;     // local transmittance product over this lane's samples
    float accl = 0.f;   // local weighted feature sum (with local transmittance)
    const int pbase = lane << 3;

#pragma unroll 2
    for (int s = 0; s < SPL; ++s) {
        const float d = 2.f + (4.f / 255.f) * (float)(pbase + s);
        const float fx = Ax + d * Bx;
        const float fy = Ay + d * By;
        const float fz = Az + d * Bz;
        float feat, den;
        tri_sample(img, opa, fx, fy, fz, feat, den);
        den *= 0.1f;                       // density scaling factor
        accl = fmaf(feat * den, Tl, accl); // weight = den * transmittance
        Tl *= (1.0f + 1e-10f) - den;       // EA raymarcher transmittance
    }

    // inclusive scan (product) of Tl across lanes
    float inc = Tl;
    for (int off = 1; off < 32; off <<= 1) {
        float t = __shfl_up(inc, off, 32);
        if (lane >= off) inc *= t;
    }
    // exclusive prefix product = transmittance entering this lane's segment
    float excl = __shfl_up(inc, 1, 32);
    if (lane == 0) excl = 1.f;

    float contrib = excl * accl;
    for (int off = 16; off; off >>= 1)
        contrib += __shfl_down(contrib, off, 32);

    // all 3 feature channels are identical -> gray == composited feature
    if (lane == 0) gray[ray] = contrib;
}

// Single wave32. Row-sum reduction via V_WMMA_F32_16X16X4_F32 with B = ones:
// D[m,n] = sum_k A[m,k] + C[m,n]  ->  accumulating C gives total sum / sumsq
// in full f32 precision. Min/max ride along as scalar lane ops.
__global__ void __launch_bounds__(32)
dvr_stats_kernel(const float* __restrict__ g, float* __restrict__ stats)
{
    const int lane = threadIdx.x;
    v2f ones; ones.x = 1.f; ones.y = 1.f;
    v8f cs = {};  // running sum accumulator
    v8f cq = {};  // running sum-of-squares accumulator
    float mn = 3.402823466e38f, mx = -3.402823466e38f;

    for (int t = 0; t < NPIX; t += 64) {     // 64 values per WMMA pair
        const float a0 = g[t + 2 * lane];
        const float a1 = g[t + 2 * lane + 1];
        mn = fminf(mn, fminf(a0, a1));
        mx = fmaxf(mx, fmaxf(a0, a1));
        v2f A;  A.x  = a0;      A.y  = a1;
        v2f A2; A2.x = a0 * a0; A2.y = a1 * a1;
        cs = __builtin_amdgcn_wmma_f32_16x16x4_f32(false, A,  false, ones,
                                                   (short)0, cs, false, false);
        cq = __builtin_amdgcn_wmma_f32_16x16x4_f32(false, A2, false, ones,
                                                   (short)0, cq, false, false);
    }

    // per-lane: sum rows held by this lane's column; total = col0 halves
    float s = cs[0] + cs[1] + cs[2] + cs[3] + cs[4] + cs[5] + cs[6] + cs[7];
    float q = cq[0] + cq[1] + cq[2] + cq[3] + cq[4] + cq[5] + cq[6] + cq[7];
    const float sum = __shfl(s, 0, 32) + __shfl(s, 16, 32); // M=0..7 + M=8..15
    const float ssq = __shfl(q, 0, 32) + __shfl(q, 16, 32);

    for (int off = 16; off; off >>= 1) {
        mn = fminf(mn, __shfl_down(mn, off, 32));
        mx = fmaxf(mx, __shfl_down(mx, off, 32));
    }

    if (lane == 0) {
        stats[0] = sum;
        stats[1] = ssq;
        stats[2] = mn;
        stats[3] = mx;
    }
}

__global__ void __launch_bounds__(256)
dvr_finalize_kernel(const float* __restrict__ g, const float* __restrict__ stats,
                    float* __restrict__ out)
{
    const int i = blockIdx.x * blockDim.x + threadIdx.x;
    const float N = (float)NPIX;
    const float sum = stats[0], ssq = stats[1], mn = stats[2], mx = stats[3];
    const float mean = sum / N;
    float var = (ssq - sum * sum / N) / (N - 1.f);   // ddof = 1
    var = fmaxf(var, 0.f);
    const float s = sqrtf(var) + 1e-8f;
    const float st    = (g[i] - mean) / s;
    const float stmin = (mn   - mean) / s;
    const float stmax = (mx   - mean) / s;
    out[i] = (st - stmin + 1e-8f) / (stmax - stmin + 1e-8f);
}

extern "C" void kernel_launch(void* const* d_in, const int* in_sizes, int n_in,
                              void* d_out, int out_size, void* d_ws, size_t ws_size,
                              hipStream_t stream)
{
    (void)in_sizes; (void)n_in; (void)out_size; (void)ws_size;
    const float* img = (const float*)d_in[0];   // [1,1,128,128,128]
    const float* opa = (const float*)d_in[1];   // [1,1,128,128,128]
    const float* R   = (const float*)d_in[2];   // [1,3,3]
    const float* T   = (const float*)d_in[3];   // [1,3]
    float* out   = (float*)d_out;               // [1,1,128,128] flat, idx = w*128+h
    float* gray  = (float*)d_ws;                // NPIX floats
    float* stats = gray + NPIX;                 // 4 floats

    // 8 rays (waves) per 256-thread block -> 2048 blocks, 16384 waves total
    dvr_render_kernel<<<NPIX / 8, 256, 0, stream>>>(img, opa, R, T, gray);
    dvr_stats_kernel<<<1, 32, 0, stream>>>(gray, stats);
    dvr_finalize_kernel<<<NPIX / 256, 256, 0, stream>>>(gray, stats, out);
}